// GNN_38920993636553
// MI455X (gfx1250) — compile-verified
//
#include <hip/hip_runtime.h>
#include <hip/hip_bf16.h>
#include <math.h>

// GAT (2 layers, H=2 heads, D=64) for MI455X / gfx1250, wave32.
// Strategy:
//  - CSR-by-destination built on device once per launch (no float atomics).
//  - feat = h @ W via V_WMMA_F32_16X16X4_F32 (full f32 precision matrix pipe).
//  - One wave per destination node: segmented softmax + aggregation in regs.

#define D_FEAT 64
#define HD 128   // H * D

typedef __attribute__((ext_vector_type(2))) float v2f;
typedef __attribute__((ext_vector_type(8))) float v8f;

// ---------------- CSR construction ----------------

__global__ __launch_bounds__(256) void count_deg_kernel(const int* __restrict__ dst,
                                                        int* __restrict__ deg, int e) {
  int i = blockIdx.x * 256 + threadIdx.x;
  if (i < e) atomicAdd(&deg[dst[i]], 1);
}

__global__ __launch_bounds__(1024) void scan_kernel(const int* __restrict__ deg,
                                                    int* __restrict__ offs,
                                                    int* __restrict__ cursor, int n) {
  __shared__ int sm[1024];
  __shared__ int carry_s;
  int t = threadIdx.x;
  if (t == 0) carry_s = 0;
  __syncthreads();
  for (int base = 0; base < n; base += 1024) {
    int x = (base + t < n) ? deg[base + t] : 0;
    sm[t] = x;
    __syncthreads();
    for (int d = 1; d < 1024; d <<= 1) {
      int v = (t >= d) ? sm[t - d] : 0;
      __syncthreads();
      sm[t] += v;
      __syncthreads();
    }
    int incl = sm[t];
    int carry = carry_s;
    if (base + t < n) {
      int ex = carry + incl - x;  // exclusive
      offs[base + t] = ex;
      cursor[base + t] = ex;
    }
    __syncthreads();
    if (t == 1023) carry_s = carry + incl;
    __syncthreads();
  }
  if (t == 0) offs[n] = carry_s;
}

__global__ __launch_bounds__(256) void scatter_csr_kernel(const int* __restrict__ src,
                                                          const int* __restrict__ dst,
                                                          int* __restrict__ cursor,
                                                          int* __restrict__ csr_src, int e) {
  int i = blockIdx.x * 256 + threadIdx.x;
  if (i < e) {
    int p = atomicAdd(&cursor[dst[i]], 1);
    csr_src[p] = src[i];
  }
}

// ---------------- GEMM: feat[N,128] = h[N,64] @ W[64,128] (f32 WMMA) ----------------
// Block = 256 threads = 8 waves; block owns 16 rows, wave w owns cols [16w,16w+16).
// Requires N % 16 == 0 (50000 = 3125*16).

__global__ __launch_bounds__(256) void gemm_feat_kernel(const float* __restrict__ h,
                                                        const float* __restrict__ W,
                                                        float* __restrict__ feat) {
  const int wave = threadIdx.x >> 5;
  const int lane = threadIdx.x & 31;
  const int row0 = blockIdx.x * 16;
  const int m = lane & 15;
  const int hl = lane >> 4;            // which half of the wave
  const int col = wave * 16 + m;       // output column
  const float* arow = h + (row0 + m) * D_FEAT + 2 * hl;
  v8f c = {};
#pragma unroll
  for (int k0 = 0; k0 < D_FEAT; k0 += 4) {
    v2f a, b;
    // A 16x4 layout: VGPR0 = K (lanes0-15) / K+2 (lanes16-31); VGPR1 = K+1 / K+3
    a.x = arow[k0];
    a.y = arow[k0 + 1];
    // B 4x16 layout mirrored: VGPR0 = row K / K+2 ; VGPR1 = row K+1 / K+3
    int kb = k0 + 2 * hl;
    b.x = W[kb * HD + col];
    b.y = W[(kb + 1) * HD + col];
    c = __builtin_amdgcn_wmma_f32_16x16x4_f32(false, a, false, b, (short)0, c,
                                              false, false);
  }
#pragma unroll
  for (int r = 0; r < 8; ++r) {
    int row = row0 + r + 8 * hl;       // C/D: VGPR r -> M=r (lanes0-15), M=r+8 (lanes16-31)
    feat[row * HD + col] = c[r];
  }
}

// ---------------- el/er: per (node, head) attention dots ----------------
// One wave per (node, head); lane covers dims {lane, lane+32}.

__global__ __launch_bounds__(256) void el_er_kernel(const float* __restrict__ feat,
                                                    const float* __restrict__ al,
                                                    const float* __restrict__ ar,
                                                    float* __restrict__ el,
                                                    float* __restrict__ er, int n) {
  int gw = (blockIdx.x * 256 + threadIdx.x) >> 5;
  int lane = threadIdx.x & 31;
  int node = gw >> 1, head = gw & 1;
  if (node >= n) return;
  const float* f = feat + node * HD + head * D_FEAT;
  const float* pl = al + head * D_FEAT;
  const float* pr = ar + head * D_FEAT;
  float f0 = f[lane], f1 = f[lane + 32];
  float vl = f0 * pl[lane] + f1 * pl[lane + 32];
  float vr = f0 * pr[lane] + f1 * pr[lane + 32];
#pragma unroll
  for (int o = 16; o > 0; o >>= 1) {
    vl += __shfl_xor(vl, o, 32);
    vr += __shfl_xor(vr, o, 32);
  }
  if (lane == 0) {
    el[node * 2 + head] = vl;
    er[node * 2 + head] = vr;
  }
}

// ---------------- Segmented softmax + aggregation: one wave per dst node ----------------

__device__ __forceinline__ float lrelu(float x) { return x > 0.0f ? x : 0.2f * x; }

__global__ __launch_bounds__(256) void gat_aggregate_kernel(
    const float* __restrict__ feat, const float* __restrict__ el,
    const float* __restrict__ er, const int* __restrict__ offs,
    const int* __restrict__ csr_src, const float* __restrict__ bias,
    float* __restrict__ out, int n) {
  int node = (blockIdx.x * 256 + threadIdx.x) >> 5;
  int lane = threadIdx.x & 31;
  if (node >= n) return;
  int jb = offs[node], je = offs[node + 1];
  int deg = je - jb;
  float er0 = er[node * 2], er1 = er[node * 2 + 1];
  const float2* el2 = (const float2*)el;

  // pass 1: segment max (lanes stride the edge list, then butterfly-reduce)
  float m0 = -__builtin_inff(), m1 = -__builtin_inff();
  for (int j = jb + lane; j < je; j += 32) {
    float2 ev = el2[csr_src[j]];
    m0 = fmaxf(m0, lrelu(ev.x + er0));
    m1 = fmaxf(m1, lrelu(ev.y + er1));
  }
#pragma unroll
  for (int o = 16; o > 0; o >>= 1) {
    m0 = fmaxf(m0, __shfl_xor(m0, o, 32));
    m1 = fmaxf(m1, __shfl_xor(m1, o, 32));
  }

  // pass 2: exp-weights, sums, and 128-dim accumulation (4 dims per lane)
  float s0 = 0.0f, s1 = 0.0f;
  float a00 = 0.0f, a01 = 0.0f, a10 = 0.0f, a11 = 0.0f;
  int s = (deg > 0) ? csr_src[jb] : 0;
  for (int j = jb; j < je; ++j) {
    int snext = (j + 1 < je) ? csr_src[j + 1] : s;
    const float* fb = feat + (long)s * HD;
    __builtin_prefetch(feat + (long)snext * HD + lane, 0, 0);  // global_prefetch_b8
    float2 ev = el2[s];
    float w0 = __expf(lrelu(ev.x + er0) - m0);
    float w1 = __expf(lrelu(ev.y + er1) - m1);
    s0 += w0;
    s1 += w1;
    a00 += w0 * fb[lane];       // head0, dim lane
    a01 += w0 * fb[lane + 32];  // head0, dim lane+32
    a10 += w1 * fb[lane + 64];  // head1, dim lane
    a11 += w1 * fb[lane + 96];  // head1, dim lane+32
    s = snext;
  }

  float inv0 = (deg > 0) ? 1.0f / s0 : 0.0f;
  float inv1 = (deg > 0) ? 1.0f / s1 : 0.0f;
  // + bias, mean over heads, ReLU
  float o0 = 0.5f * ((a00 * inv0 + bias[lane]) + (a10 * inv1 + bias[64 + lane]));
  float o1 = 0.5f * ((a01 * inv0 + bias[32 + lane]) + (a11 * inv1 + bias[96 + lane]));
  out[node * D_FEAT + lane] = o0 > 0.0f ? o0 : 0.0f;
  out[node * D_FEAT + 32 + lane] = o1 > 0.0f ? o1 : 0.0f;
}

// ---------------- host-side launch ----------------

static inline char* align256(char* p) {
  return (char*)(((uintptr_t)p + 255) & ~(uintptr_t)255);
}

extern "C" void kernel_launch(void* const* d_in, const int* in_sizes, int n_in,
                              void* d_out, int out_size, void* d_ws, size_t ws_size,
                              hipStream_t stream) {
  const float* x  = (const float*)d_in[0];
  const float* W1 = (const float*)d_in[1];
  const float* al1 = (const float*)d_in[2];
  const float* ar1 = (const float*)d_in[3];
  const float* b1 = (const float*)d_in[4];
  const float* W2 = (const float*)d_in[5];
  const float* al2 = (const float*)d_in[6];
  const float* ar2 = (const float*)d_in[7];
  const float* b2 = (const float*)d_in[8];
  const int* src = (const int*)d_in[9];
  const int* dst = (const int*)d_in[10];

  const int n = in_sizes[0] / D_FEAT;  // 50000 (multiple of 16)
  const int e = in_sizes[9];           // 800000

  // workspace partition
  char* w = (char*)d_ws;
  float* feat = (float*)w;           w = align256(w + (size_t)n * HD * 4);
  float* h1 = (float*)w;             w = align256(w + (size_t)n * D_FEAT * 4);
  float* el = (float*)w;             w = align256(w + (size_t)n * 2 * 4);
  float* er = (float*)w;             w = align256(w + (size_t)n * 2 * 4);
  int* deg = (int*)w;                w = align256(w + (size_t)n * 4);
  int* offs = (int*)w;               w = align256(w + (size_t)(n + 1) * 4);
  int* cursor = (int*)w;             w = align256(w + (size_t)n * 4);
  int* csr_src = (int*)w;            w = align256(w + (size_t)e * 4);

  const int eb = (e + 255) / 256;
  const int gemm_blocks = n / 16;             // 3125
  const int eler_blocks = (n * 2 * 32 + 255) / 256;  // waves = 2n
  const int agg_blocks = (n * 32 + 255) / 256;       // waves = n

  // CSR by destination (shared across both layers)
  hipMemsetAsync(deg, 0, (size_t)n * 4, stream);
  count_deg_kernel<<<eb, 256, 0, stream>>>(dst, deg, e);
  scan_kernel<<<1, 1024, 0, stream>>>(deg, offs, cursor, n);
  scatter_csr_kernel<<<eb, 256, 0, stream>>>(src, dst, cursor, csr_src, e);

  // layer 1
  gemm_feat_kernel<<<gemm_blocks, 256, 0, stream>>>(x, W1, feat);
  el_er_kernel<<<eler_blocks, 256, 0, stream>>>(feat, al1, ar1, el, er, n);
  gat_aggregate_kernel<<<agg_blocks, 256, 0, stream>>>(feat, el, er, offs, csr_src,
                                                       b1, h1, n);
  // layer 2
  gemm_feat_kernel<<<gemm_blocks, 256, 0, stream>>>(h1, W2, feat);
  el_er_kernel<<<eler_blocks, 256, 0, stream>>>(feat, al2, ar2, el, er, n);
  gat_aggregate_kernel<<<agg_blocks, 256, 0, stream>>>(feat, el, er, offs, csr_src,
                                                       b2, (float*)d_out, n);
}